// BatchPitNorm1d_43336220017297
// MI455X (gfx1250) — compile-verified
//
#include <hip/hip_runtime.h>
#include <cstdint>
#include <cstddef>

// Problem shape (fixed by the reference).
#define Bsz 512
#define Ssz 2048
#define Fsz 128

// Tiling: 4 batch rows per block, 2 rows per thread (256 thr = 2 x 128 lanes),
// cdf_data staged through LDS in 32-row tiles (32*128*4B = 16 KB, x2 buffers).
#define BT  4
#define RPT 2
#define ST  32
#define NT  (Ssz / ST)

#if defined(__AMDGCN__) && __has_builtin(__builtin_amdgcn_tensor_load_to_lds)
#define USE_TDM 1
#else
#define USE_TDM 0
#endif

typedef unsigned int v4u __attribute__((ext_vector_type(4)));
typedef int          v4i __attribute__((ext_vector_type(4)));
typedef int          v8i __attribute__((ext_vector_type(8)));

// Issue one TDM DMA: 2D tile (ST rows x 128 cols, f32) of cdf_data -> LDS.
// Called by a single (uniform-branch) wave per workgroup; TDM ignores EXEC.
__device__ __forceinline__ void tile_load_issue(const float* gsrc, float* ldst) {
#if USE_TDM
  const unsigned long long ga  = (unsigned long long)(uintptr_t)gsrc;
  const unsigned int       lds = (unsigned int)(uintptr_t)ldst; // low 32 bits = LDS offset

  // D# group 0: [1:0] count=1 (valid user desc), [63:32] lds_addr,
  //             [120:64] global_addr, [127:126] type=2 ("image").
  v4u g0 = { 1u,
             lds,
             (unsigned int)(ga & 0xffffffffu),
             (unsigned int)((ga >> 32) & 0x01ffffffu) | (2u << 30) };

  // D# group 1: [17:16] data_size=2 (4B), [79:48] tensor_dim0=128,
  //             [111:80] tensor_dim1=2048, [127:112] tile_dim0=128,
  //             [143:128] tile_dim1=ST, [207:160] tensor_dim0_stride=128.
  v8i g1 = { (int)(2u << 16),
             (int)(128u << 16),
             (int)(2048u << 16),
             (int)(128u << 16),
             (int)ST,
             (int)128,
             0, 0 };

  v4i gz = { 0, 0, 0, 0 };
#if __clang_major__ >= 23
  v8i gz8 = { 0, 0, 0, 0, 0, 0, 0, 0 };
  __builtin_amdgcn_tensor_load_to_lds(g0, g1, gz, gz, gz8, 0);
#else
  __builtin_amdgcn_tensor_load_to_lds(g0, g1, gz, gz, 0);
#endif
#else
  // Fallback (host parse / missing builtin): synchronous copy by calling wave.
  for (int i = (int)(threadIdx.x & 31); i < ST * Fsz; i += 32) ldst[i] = gsrc[i];
#endif
}

__device__ __forceinline__ void tile_wait_keep(int keep_in_flight) {
#if USE_TDM
  if (keep_in_flight) __builtin_amdgcn_s_wait_tensorcnt(1);
  else                __builtin_amdgcn_s_wait_tensorcnt(0);
#else
  (void)keep_in_flight;
#endif
}

// Branch-free Phi given the pre-scaled erfc argument a = (c - x)/(bw*sqrt2):
//   Phi((x-c)/bw) = 0.5 * erfc(a)
// Numerical-Recipes rational erfc: t = 1/(1 + 0.5|a|),
//   erfc(|a|) = t * exp(-a^2 + P9(t)),  erfc(a<0) = 2 - erfc(|a|).
// ~16 VALU ops: v_rcp + v_exp (TRANS, co-executes with FMAs) + 11 FMA + selects.
__device__ __forceinline__ float phi_from_arg(float a) {
  const float aa = __builtin_fabsf(a);
#if defined(__AMDGCN__)
  const float t = __builtin_amdgcn_rcpf(__builtin_fmaf(0.5f, aa, 1.0f));
#else
  const float t = 1.0f / (0.5f * aa + 1.0f);
#endif
  float p = 0.17087277f;
  p = __builtin_fmaf(p, t, -0.82215223f);
  p = __builtin_fmaf(p, t,  1.48851587f);
  p = __builtin_fmaf(p, t, -1.13520398f);
  p = __builtin_fmaf(p, t,  0.27886807f);
  p = __builtin_fmaf(p, t, -0.18628806f);
  p = __builtin_fmaf(p, t,  0.09678418f);
  p = __builtin_fmaf(p, t,  0.37409196f);
  p = __builtin_fmaf(p, t,  1.00002368f);
  p = __builtin_fmaf(p, t, -1.26551223f);
  const float e  = __expf(__builtin_fmaf(-aa, aa, p));   // v_exp_f32 path
  const float ep = t * e;                                // erfc(|a|)
  const float ec = (a < 0.0f) ? (2.0f - ep) : ep;        // v_cndmask
  return 0.5f * ec;
}

__device__ __forceinline__ float phi_inv(float u) { return normcdfinvf(u); }

__global__ __launch_bounds__(256) void batch_pit_norm1d_kernel(
    const float* __restrict__ x,        // [Bsz, Fsz]
    const float* __restrict__ cdf,      // [Ssz, Fsz]
    const float* __restrict__ bwp,      // [1, Fsz]
    float* __restrict__ out) {          // [Bsz, Fsz]
  __shared__ float lbuf[2][ST * Fsz];   // 32 KB double buffer

  const int  tid    = (int)threadIdx.x;
  const int  f      = tid & (Fsz - 1);  // feature lane
  const int  bh     = tid >> 7;         // 0..1
  const int  b0     = (int)blockIdx.x * BT + bh * RPT;
  const bool issuer = (tid < 32);       // wave 0 drives the TDM pipeline

  // bw = sigmoid(p) => 1/bw = 1 + exp(-p).  Fold 1/sqrt(2) in as well so the
  // erfc argument is a single subtract:  a = c*k - x*k,  k = (1+e^-p)/sqrt2.
  const float k = (1.0f + __expf(-bwp[f])) * 0.70710678118654752f;

  float xs[RPT], acc[RPT];
#pragma unroll
  for (int r = 0; r < RPT; ++r) {
    xs[r]  = x[(size_t)(b0 + r) * Fsz + f] * k;
    acc[r] = 0.0f;
  }

  // Prologue: DMA tile 0 into buffer 0.
  if (issuer) tile_load_issue(cdf, &lbuf[0][0]);

  for (int t = 0; t < NT; ++t) {
    const int cur = t & 1;
    if (issuer) {
      if (t + 1 < NT) {
        // Prefetch next tile into the other buffer; tile t is complete once
        // TENSORcnt <= 1 (TDM ops from one wave retire in order).
        tile_load_issue(cdf + (size_t)(t + 1) * ST * Fsz, &lbuf[cur ^ 1][0]);
        tile_wait_keep(1);
      } else {
        tile_wait_keep(0);
      }
    }
    __syncthreads();  // tile t visible to all waves

    const float* __restrict__ cp = &lbuf[cur][0];
#pragma unroll 8
    for (int s = 0; s < ST; ++s) {
      const float cs = cp[s * Fsz + f] * k;   // LDS read amortized over RPT rows
#pragma unroll
      for (int r = 0; r < RPT; ++r) acc[r] += phi_from_arg(cs - xs[r]);
    }
    __syncthreads();  // compute done before this buffer is overwritten (t+2)
  }

  const float scale = 1.0f / (float)Ssz;
#pragma unroll
  for (int r = 0; r < RPT; ++r)
    out[(size_t)(b0 + r) * Fsz + f] = phi_inv(acc[r] * scale);
}

extern "C" void kernel_launch(void* const* d_in, const int* in_sizes, int n_in,
                              void* d_out, int out_size, void* d_ws, size_t ws_size,
                              hipStream_t stream) {
  (void)in_sizes; (void)n_in; (void)d_ws; (void)ws_size; (void)out_size;
  const float* x   = (const float*)d_in[0];  // [512,128]
  const float* cdf = (const float*)d_in[1];  // [2048,128]
  const float* bwp = (const float*)d_in[2];  // [1,128]
  float*       out = (float*)d_out;          // [512,128]

  dim3 grid(Bsz / BT);   // 128 blocks
  dim3 block(256);       // 8 wave32 waves
  hipLaunchKernelGGL(batch_pit_norm1d_kernel, grid, block, 0, stream,
                     x, cdf, bwp, out);
}